// RNNEncoder_40724879901195
// MI455X (gfx1250) — compile-verified
//
#include <hip/hip_runtime.h>

// GRU cell step: memory-bound GEMV pair (252 MB fp32 weights @ 23.3 TB/s ~= 10.8us floor).
// GEMVs are expressed as V_WMMA_F32_16X16X4_F32 with broadcast-B so the weight stream
// feeds the matrix engine directly at no extra bandwidth cost.

typedef __attribute__((ext_vector_type(2))) float v2f;
typedef __attribute__((ext_vector_type(4))) float v4f;
typedef __attribute__((ext_vector_type(8))) float v8f;

#define HDIM 4096
#define EDIM 1024
#define GDIM 6
#define KIN  1030          // E + G
#define KINP 1032          // padded to multiple of 4 (zeros)
#define ROWS (3 * HDIM)    // 12288 gate pre-activations per side
#define WAVES_PB 8         // 256 threads / wave32
#define ROWS_PER_WAVE 16
#define ROWS_PB (WAVES_PB * ROWS_PER_WAVE)   // 128 rows per block
#define BLOCKS_PER_MAT (ROWS / ROWS_PB)      // 96

// ---------------------------------------------------------------- phase 1: x = [emb[tok]; goals; 0,0]
__global__ void build_x_kernel(const int* __restrict__ tok,
                               const float* __restrict__ emb,
                               const float* __restrict__ goals,
                               float* __restrict__ x) {
    int i = blockIdx.x * blockDim.x + threadIdx.x;
    if (i >= KINP) return;
    int t = tok[0];                       // int64 input: low 32 bits hold value < 32000
    float v = 0.0f;
    if (i < EDIM)       v = emb[(size_t)t * EDIM + i];
    else if (i < KIN)   v = goals[i - EDIM];
    x[i] = v;                             // i in [KIN, KINP) -> zero pad
}

// ---------------------------------------------------------------- phase 2: fused dual GEMV via WMMA
// blocks [0,96)  : gi = W_ih @ x + b_ih   (K = 1030, padded loop to 1032)
// blocks [96,192): gh = W_hh @ h + b_hh   (K = 4096)
__global__ __launch_bounds__(256) void gru_gemv_wmma(
    const float* __restrict__ W_ih, const float* __restrict__ b_ih,
    const float* __restrict__ xv,
    const float* __restrict__ W_hh, const float* __restrict__ b_hh,
    const float* __restrict__ hprev,
    float* __restrict__ gi, float* __restrict__ gh) {

    __shared__ float xs[HDIM];

    const float* W; const float* bias; const float* src; float* out;
    int K, Kloop, mb;
    if (blockIdx.x < BLOCKS_PER_MAT) {
        W = W_ih; bias = b_ih; src = xv;    out = gi; K = KIN;  Kloop = KINP; mb = blockIdx.x;
    } else {
        W = W_hh; bias = b_hh; src = hprev; out = gh; K = HDIM; Kloop = HDIM; mb = blockIdx.x - BLOCKS_PER_MAT;
    }

    // Stage the shared vector in LDS (zero-padded for the W_ih tail tile).
    for (int i = threadIdx.x; i < Kloop; i += blockDim.x)
        xs[i] = (i < K) ? src[i] : 0.0f;
    __syncthreads();

    const int lane = threadIdx.x & 31;
    const int wave = threadIdx.x >> 5;
    const int row0 = (mb * WAVES_PB + wave) * ROWS_PER_WAVE;
    const int m    = lane & 15;           // A-matrix row within tile (lanes 0-15 / 16-31)
    const int lh   = lane >> 4;           // lane-half: selects K-slot pair {0,1} vs {2,3}
    const float* wrow = W + (size_t)(row0 + m) * (size_t)K;

    v8f c = {0.f, 0.f, 0.f, 0.f, 0.f, 0.f, 0.f, 0.f};

    if (Kloop == HDIM) {
        // W_hh: rows are 16KB (16B aligned) -> B128 per lane covers 8 K-values, 2 WMMAs per load.
        // K-slot permutation: WMMA0 contracts {k,k+1 (lo-half), k+4,k+5 (hi-half)},
        //                     WMMA1 contracts {k+2,k+3, k+6,k+7}; A and B use the same mapping.
        #pragma unroll 2
        for (int k = 0; k < HDIM; k += 8) {
            const int kb = k + lh * 4;
            v4f a = *(const v4f*)(wrow + kb);
            v4f b = *(const v4f*)(xs + kb);
            c = __builtin_amdgcn_wmma_f32_16x16x4_f32(false, a.lo, false, b.lo, (short)0, c, false, false);
            c = __builtin_amdgcn_wmma_f32_16x16x4_f32(false, a.hi, false, b.hi, (short)0, c, false, false);
        }
    } else {
        // W_ih: rows of 1030 floats are only 8B aligned -> B64 per lane, step 4.
        const int kmain = K & ~3;         // 1028
        #pragma unroll 4
        for (int k = 0; k < kmain; k += 4) {
            const int ka = k + lh * 2;
            v2f a = *(const v2f*)(wrow + ka);
            v2f b = *(const v2f*)(xs + ka);
            c = __builtin_amdgcn_wmma_f32_16x16x4_f32(false, a, false, b, (short)0, c, false, false);
        }
        // Tail tile: hi-half lanes would read K=1030,1031 -> clamp W address (x pad is zero,
        // so the product contributes nothing); lo-half lanes cover the real K=1028,1029.
        for (int k = kmain; k < Kloop; k += 4) {
            const int ka = k + lh * 2;
            int kw = ka; if (kw > K - 2) kw = K - 2;
            v2f a = *(const v2f*)(wrow + kw);
            v2f b = *(const v2f*)(xs + ka);
            c = __builtin_amdgcn_wmma_f32_16x16x4_f32(false, a, false, b, (short)0, c, false, false);
        }
    }

    // B is column-broadcast, so D[m][n] == y[m] for every n.
    // C/D layout: VGPR j = row j (lanes 0-15) / row j+8 (lanes 16-31).
    if (m == 0) {
        const int rb = row0 + lh * 8;
        #pragma unroll
        for (int j = 0; j < 8; ++j)
            out[rb + j] = c[j] + bias[rb + j];
    }
}

// ---------------------------------------------------------------- phase 3: gate combine
__global__ void gru_gates(const float* __restrict__ gi, const float* __restrict__ gh,
                          const float* __restrict__ hprev, float* __restrict__ hout) {
    int i = blockIdx.x * blockDim.x + threadIdx.x;
    if (i >= HDIM) return;
    float r = 1.0f / (1.0f + __expf(-(gi[i]            + gh[i])));
    float z = 1.0f / (1.0f + __expf(-(gi[HDIM + i]     + gh[HDIM + i])));
    float n = tanhf(gi[2 * HDIM + i] + r * gh[2 * HDIM + i]);
    hout[i] = (1.0f - z) * n + z * hprev[i];
}

// ---------------------------------------------------------------- launcher
extern "C" void kernel_launch(void* const* d_in, const int* in_sizes, int n_in,
                              void* d_out, int out_size, void* d_ws, size_t ws_size,
                              hipStream_t stream) {
    (void)in_sizes; (void)n_in; (void)out_size; (void)ws_size;
    const int*   tok   = (const int*)d_in[0];
    const float* hprev = (const float*)d_in[1];
    const float* goals = (const float*)d_in[2];
    const float* emb   = (const float*)d_in[3];
    const float* W_ih  = (const float*)d_in[4];
    const float* b_ih  = (const float*)d_in[5];
    const float* W_hh  = (const float*)d_in[6];
    const float* b_hh  = (const float*)d_in[7];
    float* out = (float*)d_out;

    // workspace layout (floats): [x: 1032 (pad to 1088)] [gi: 12288] [gh: 12288]  ~100 KB
    float* x  = (float*)d_ws;
    float* gi = x + 1088;
    float* gh = gi + ROWS;

    build_x_kernel<<<(KINP + 255) / 256, 256, 0, stream>>>(tok, emb, goals, x);
    gru_gemv_wmma<<<2 * BLOCKS_PER_MAT, 256, 0, stream>>>(W_ih, b_ih, x, W_hh, b_hh, hprev, gi, gh);
    gru_gates<<<(HDIM + 255) / 256, 256, 0, stream>>>(gi, gh, hprev, out);
}